// GrammarTree20PDE_32469952757802
// MI455X (gfx1250) — compile-verified
//
#include <hip/hip_runtime.h>

#ifndef __has_builtin
#define __has_builtin(x) 0
#endif

// ---------------- problem constants (match reference) ----------------
#define NXG     131072        // spatial points
#define NSTEPS  128           // sequential Euler steps

constexpr int THREADS = 256;                 // 8 wave32s per workgroup
constexpr int PTS     = 3;                   // points per thread (registers)
constexpr int HALO    = NSTEPS;              // halo width = #steps (radius-1 stencil)
constexpr int COMP    = THREADS * PTS;       // 768 computed points per block
constexpr int TILE    = COMP - 2 * HALO;     // 512 interior points per block
constexpr int LTOT    = COMP + 2;            // 770 incl. guard cells
constexpr int NBLK    = NXG / TILE;          // 256 blocks
static_assert(TILE > 0 && NXG % TILE == 0, "tiling must divide NX");
static_assert(NBLK == THREADS, "stats kernel assumes NBLK == THREADS");

// ---------------- CDNA5 async global->LDS staging ----------------
__device__ __forceinline__ void async_load_f32_to_lds(float* lds_dst, const float* gsrc) {
#if __has_builtin(__builtin_amdgcn_global_load_async_to_lds_b32)
  __builtin_amdgcn_global_load_async_to_lds_b32(
      (__attribute__((address_space(1))) int*)(void*)gsrc,
      (__attribute__((address_space(3))) int*)(void*)lds_dst,
      0, 0);
#else
  unsigned lds_off =
      (unsigned)(unsigned long long)(__attribute__((address_space(3))) float*)lds_dst;
  asm volatile("global_load_async_to_lds_b32 %0, %1, off"
               :: "v"(lds_off), "v"(gsrc) : "memory");
#endif
}

__device__ __forceinline__ void wait_asynccnt0() {
#if __has_builtin(__builtin_amdgcn_s_wait_asynccnt)
  __builtin_amdgcn_s_wait_asynccnt(0);
#else
  asm volatile("s_wait_asynccnt 0" ::: "memory");
#endif
}

// ---------------- per-point PDE update ----------------
// DX=0.1 -> 1/(2DX)=5.0 (exact), 1/DX^2=100.0 ; DT=1e-3
__device__ __forceinline__ float pde_point(float uL, float uC, float uR,
                                           const float* wr, float gam, int g) {
  float u_x  = (uR - uL) * 5.0f;
  float u_xx = (uR - 2.0f * uC + uL) * 100.0f;
  float u2 = uC * uC;
  float u3 = u2 * uC;
  float r = wr[0] * uC + wr[1] * u_x + wr[2] * u_xx + wr[3] * u2 + wr[4] * u3
          + wr[5] * (uC * u_x) + wr[6] * (uC * u_xx) + wr[7] * (u_x * u_xx);
  float rhs = 0.8f * u_xx + 0.5f * uC - u3 + gam * r;
  float un = uC + 1e-3f * rhs;
  // Dirichlet firewall each step
  return (g == 0 || g == NXG - 1) ? 0.0f : un;
}

// ---------------- kernel 1: all 128 steps, halo-blocked ----------------
__global__ __launch_bounds__(THREADS)
void pde_kernel(const float* __restrict__ u0, const float* __restrict__ w,
                const float* __restrict__ gamma, float* __restrict__ uout,
                float* __restrict__ partial) {
  __shared__ float tile[LTOT];
  __shared__ float shFirst[2][THREADS];
  __shared__ float shLast[2][THREADS];
  __shared__ float red[THREADS];

  const int tid  = threadIdx.x;
  const int blk  = blockIdx.x;
  const int base = blk * TILE - HALO - 1;   // global index g = base + local j

  // Stage extended tile into LDS via async DMA path (OOB -> 0)
  for (int j = tid; j < LTOT; j += THREADS) {
    int g = base + j;
    if (g >= 0 && g < NXG) {
      async_load_f32_to_lds(&tile[j], u0 + g);
    } else {
      tile[j] = 0.0f;
    }
  }
  wait_asynccnt0();
  __syncthreads();

  float wr[8];
#pragma unroll
  for (int i = 0; i < 8; ++i) wr[i] = w[i];
  const float gam = gamma[0];

  const int j0 = 1 + PTS * tid;
  const int g0 = base + j0;
  float r0 = tile[j0], r1 = tile[j0 + 1], r2 = tile[j0 + 2];
  const float gL = tile[0];          // guard cells (containment only)
  const float gR = tile[LTOT - 1];

  int buf = 0;
  for (int s = 0; s < NSTEPS; ++s) {
    shFirst[buf][tid] = r0;
    shLast[buf][tid]  = r2;
    __syncthreads();
    float left  = (tid == 0)           ? gL : shLast[buf][tid - 1];
    float right = (tid == THREADS - 1) ? gR : shFirst[buf][tid + 1];
    float n0 = pde_point(left, r0, r1, wr, gam, g0);
    float n1 = pde_point(r0, r1, r2, wr, gam, g0 + 1);
    float n2 = pde_point(r1, r2, right, wr, gam, g0 + 2);
    r0 = n0; r1 = n1; r2 = n2;
    buf ^= 1;
  }

  // Write interior points and accumulate block energy partial
  float esum = 0.0f;
#pragma unroll
  for (int i = 0; i < PTS; ++i) {
    int j = j0 + i;
    float v = (i == 0) ? r0 : ((i == 1) ? r1 : r2);
    if (j > HALO && j <= HALO + TILE) {
      uout[base + j] = v;
      esum += v * v;
    }
  }
  red[tid] = esum;
  __syncthreads();
  for (int off = THREADS / 2; off > 0; off >>= 1) {
    if (tid < off) red[tid] += red[tid + off];
    __syncthreads();
  }
  if (tid == 0) partial[blk] = red[0];
}

// ---------------- kernel 2: phi2 + entropy (deterministic) ----------------
__global__ __launch_bounds__(THREADS)
void stats_kernel(const float* __restrict__ u, const float* __restrict__ partial,
                  float* __restrict__ out /* -> d_out + NXG */) {
  __shared__ float sh[THREADS];
  const int tid = threadIdx.x;

  sh[tid] = partial[tid];                   // NBLK == THREADS
  __syncthreads();
  for (int off = THREADS / 2; off > 0; off >>= 1) {
    if (tid < off) sh[tid] += sh[tid + off];
    __syncthreads();
  }
  const float s = sh[0];
  __syncthreads();                          // before reusing sh

  const float sp = s + 1e-12f;
  float acc = 0.0f;
  for (int i = tid; i < NXG; i += THREADS) {
    float uu = u[i];
    float e  = uu * uu;
    float p  = e / sp;
    acc += p * logf(p + 1e-12f);
  }
  sh[tid] = acc;
  __syncthreads();
  for (int off = THREADS / 2; off > 0; off >>= 1) {
    if (tid < off) sh[tid] += sh[tid + off];
    __syncthreads();
  }
  if (tid == 0) {
    out[0] = s * 0.1f;   // phi2 = sum(u^2) * DX
    out[1] = -sh[0];     // Shannon entropy
  }
}

// ---------------- entry point ----------------
extern "C" void kernel_launch(void* const* d_in, const int* in_sizes, int n_in,
                              void* d_out, int out_size, void* d_ws, size_t ws_size,
                              hipStream_t stream) {
  const float* u0    = (const float*)d_in[0];
  const float* w     = (const float*)d_in[1];
  const float* gamma = (const float*)d_in[2];
  // d_in[3] = Nt (always 128 in this problem; hardcoded as NSTEPS)

  float* uout    = (float*)d_out;       // [0, NXG)  : final u
  float* partial = (float*)d_ws;        // NBLK block energy partials

  pde_kernel<<<NBLK, THREADS, 0, stream>>>(u0, w, gamma, uout, partial);
  stats_kernel<<<1, THREADS, 0, stream>>>(uout, partial, uout + NXG);
}